// GHCF_GLN_11905649344757
// MI455X (gfx1250) — compile-verified
//
#include <hip/hip_runtime.h>

typedef __attribute__((ext_vector_type(2))) float v2f;
typedef __attribute__((ext_vector_type(4))) float v4f;
typedef __attribute__((ext_vector_type(8))) float v8f;

#define D 128
#define W_BUY  0.5f
#define W_CART 0.3333333f
#define W_PV   0.1666667f

#if defined(__gfx1250__) && __has_builtin(__builtin_amdgcn_global_load_async_to_lds_b128)
#define HAVE_ASYNC 1
#else
#define HAVE_ASYNC 0
#endif

#if HAVE_ASYNC
typedef int v4i __attribute__((vector_size(16)));
typedef __attribute__((address_space(1))) v4i g_v4i;  // global 16B vector
typedef __attribute__((address_space(3))) v4i l_v4i;  // LDS 16B vector

__device__ __forceinline__ void async_cp16(const float* g, float* l) {
  // GLOBAL_LOAD_ASYNC_TO_LDS_B128: memory -> LDS, tracked with ASYNCcnt
  __builtin_amdgcn_global_load_async_to_lds_b128((g_v4i*)g, (l_v4i*)l, 0, 0);
}
#endif

__device__ __forceinline__ void async_wait() {
#if HAVE_ASYNC
#if __has_builtin(__builtin_amdgcn_s_wait_asynccnt)
  __builtin_amdgcn_s_wait_asynccnt(0);
#else
  asm volatile("s_wait_asynccnt 0x0" ::: "memory");
#endif
#endif
}

// Stage 64 rows x 128 cols (gathered through ids) into an LDS buffer.
// Full slabs use async-to-LDS (raw emb, no scaling); partial tail slab uses
// the predicated synchronous path with zero fill.
__device__ __forceinline__ void stage64(const float* __restrict__ emb,
                                        const int* __restrict__ ids,
                                        int n_rows, int base,
                                        float* __restrict__ buf, int tid) {
#if HAVE_ASYNC
  if (base + 64 <= n_rows) {
    for (int p = tid; p < 64 * (D / 4); p += 256) {
      const int r  = p >> 5;  // 32 float4 per row
      const int c4 = p & 31;
      async_cp16(emb + (size_t)ids[base + r] * D + c4 * 4,
                 &buf[r * D + c4 * 4]);
    }
    return;
  }
#endif
  for (int p = tid; p < 64 * (D / 4); p += 256) {
    const int r  = p >> 5;
    const int c4 = p & 31;
    v4f val = {};
    if (base + r < n_rows)
      val = *(const v4f*)(emb + (size_t)ids[base + r] * D + c4 * 4);
    *(v4f*)&buf[r * D + c4 * 4] = val;
  }
}

// ---------------------------------------------------------------------------
// Kernel 0: zero the Gram accumulators in workspace
// ---------------------------------------------------------------------------
__global__ void zero_kernel(float* __restrict__ ws, int n) {
  int i = blockIdx.x * blockDim.x + threadIdx.x;
  if (i < n) ws[i] = 0.0f;
}

// ---------------------------------------------------------------------------
// Kernel 1: G += 0.01 * emb[ids]^T emb[ids]   (= x0^T x0, 128x128 f32 Gram)
// blockDim = 256 (8 wave32). Double-buffered 64-row LDS slabs, async staged.
// Wave w owns M-block w x all 8 N-blocks -> 8 v8f accumulators.
// ---------------------------------------------------------------------------
__global__ void gram_kernel(const float* __restrict__ emb,
                            const int* __restrict__ ids,
                            int n_rows,
                            float* __restrict__ G) {
  extern __shared__ float sm[];
  float* buf0 = sm;            // 32 KB
  float* buf1 = sm + 64 * D;   // 32 KB
  const int tid  = threadIdx.x;
  const int lane = tid & 31;
  const int wave = tid >> 5;
  const int l16  = lane & 15;
  const int kg   = (lane >> 4) * 2;
  const int n_slabs = (n_rows + 63) >> 6;

  v8f acc[8] = {};

  int s = blockIdx.x;
  if (s < n_slabs) stage64(emb, ids, n_rows, s << 6, buf0, tid);
  int par = 0;
  for (; s < n_slabs; s += gridDim.x) {
    async_wait();       // own async loads for current slab complete
    __syncthreads();    // all waves' loads visible; prev buffer free for reuse
    float* cur = par ? buf1 : buf0;
    float* nxt = par ? buf0 : buf1;
    const int snext = s + (int)gridDim.x;
    if (snext < n_slabs) stage64(emb, ids, n_rows, snext << 6, nxt, tid);

    const int col_m = wave * 16 + l16;  // A = X^T tile column
    for (int k0 = 0; k0 < 64; k0 += 4) {
      v2f a;
      a.x = cur[(k0 + kg) * D + col_m];
      a.y = cur[(k0 + kg + 1) * D + col_m];
#pragma unroll
      for (int n = 0; n < 8; ++n) {
        const int col = n * 16 + l16;
        v2f b;
        b.x = cur[(k0 + kg) * D + col];
        b.y = cur[(k0 + kg + 1) * D + col];
        acc[n] = __builtin_amdgcn_wmma_f32_16x16x4_f32(
            false, a, false, b, (short)0, acc[n], false, false);
      }
    }
    par ^= 1;
  }

  // reduce into global Gram; fold A_SCALE^2 = 0.01 here (raw rows were staged)
  const int row_half = (lane >> 4) * 8;
#pragma unroll
  for (int n = 0; n < 8; ++n) {
#pragma unroll
    for (int v = 0; v < 8; ++v) {
      const int row = wave * 16 + row_half + v;
      atomicAdd(&G[row * D + n * 16 + l16], acc[n][v] * 0.01f);
    }
  }
}

// ---------------------------------------------------------------------------
// Kernel 2: tiny D-space algebra. One block, 128 threads.
// Aeff = 0.1*(0.5 I + sum_b 0.25*mtl_b*(c0_b (x) p_b + c1_b (x) q_b))
// Beff = 0.1*(0.5 I + sum_b 0.25*mtl_b*(c3_b (x) r_b + c2_b (x) s_b))
// new_es[b] = 0.5*(e @ edge_w[b] + e)
// ---------------------------------------------------------------------------
__global__ void build_kernel(const float* __restrict__ edge_emb,
                             const float* __restrict__ w_vec,
                             const float* __restrict__ w_mat,
                             const float* __restrict__ edge_w,
                             const float* __restrict__ Gx,
                             const float* __restrict__ Gy,
                             float* __restrict__ Aeff,
                             float* __restrict__ Beff,
                             float* __restrict__ es_out) {
  __shared__ float e[D];
  __shared__ float c0[D], c1[D], c2[D], c3[D];
  __shared__ float vg[4][D];
  __shared__ float p[D], q[D], r[D], sv[D];
  const int t = threadIdx.x;  // 128 threads
  const float mtl[3] = {W_BUY, W_CART, W_PV};

  e[t] = edge_emb[t] * 0.1f;  // B_SCALE
  for (int i = 0; i < D; ++i) {
    Aeff[i * D + t] = (i == t) ? 0.05f : 0.0f;
    Beff[i * D + t] = (i == t) ? 0.05f : 0.0f;
  }
  __syncthreads();

  for (int b = 0; b < 3; ++b) {
    const float* wv = w_vec + b * 4 * D;
    c0[t] = e[t] * wv[0 * D + t];  // uu
    c1[t] = e[t] * wv[1 * D + t];  // ui
    c2[t] = e[t] * wv[2 * D + t];  // iu
    c3[t] = e[t] * wv[3 * D + t];  // ii
    __syncthreads();

    float s0 = 0.f, s1 = 0.f, s2 = 0.f, s3 = 0.f;
    for (int i = 0; i < D; ++i) {
      s0 += Gx[t * D + i] * c0[i];
      s1 += Gy[t * D + i] * c1[i];
      s2 += Gx[t * D + i] * c2[i];
      s3 += Gy[t * D + i] * c3[i];
    }
    vg[0][t] = s0; vg[1][t] = s1; vg[2][t] = s2; vg[3][t] = s3;
    __syncthreads();

    const float* m0 = w_mat + (size_t)(b * 4 + 0) * D * D;
    const float* m1 = w_mat + (size_t)(b * 4 + 1) * D * D;
    const float* m2 = w_mat + (size_t)(b * 4 + 2) * D * D;
    const float* m3 = w_mat + (size_t)(b * 4 + 3) * D * D;
    float ps = 0.f, qs = 0.f, ssv = 0.f, rs = 0.f;
    for (int i = 0; i < D; ++i) {
      ps  += vg[0][i] * m0[i * D + t];  // p = (Gx c0) @ W[b,0]
      qs  += vg[1][i] * m1[i * D + t];  // q = (Gy c1) @ W[b,1]
      ssv += vg[2][i] * m2[i * D + t];  // s = (Gx c2) @ W[b,2]
      rs  += vg[3][i] * m3[i * D + t];  // r = (Gy c3) @ W[b,3]
    }
    p[t] = ps; q[t] = qs; sv[t] = ssv; r[t] = rs;
    __syncthreads();

    const float f = 0.025f * mtl[b];  // 0.1 * 0.5 * 0.5 * mtl
    for (int i = 0; i < D; ++i) {
      Aeff[i * D + t] += f * (c0[i] * p[t] + c1[i] * q[t]);
      Beff[i * D + t] += f * (c3[i] * r[t] + c2[i] * sv[t]);
    }

    const float* ew = edge_w + (size_t)b * D * D;
    float ea = 0.f;
    for (int i = 0; i < D; ++i) ea += e[i] * ew[i * D + t];
    es_out[b * D + t] = 0.5f * (ea + e[t]);
    __syncthreads();
  }
}

// ---------------------------------------------------------------------------
// Kernel 3: out = emb[ids] @ T  (T = Aeff/Beff cached in LDS; 0.1 folded in T).
// blockDim = 256 (8 waves). Double-buffered async 64-row slabs.
// Wave w: M-block (w>>1), N-half (w&1): 4 tiles x 32 K-steps of f32 WMMA.
// ---------------------------------------------------------------------------
__global__ void apply_kernel(const float* __restrict__ emb,
                             const int* __restrict__ ids,
                             int n_rows,
                             const float* __restrict__ T,
                             float* __restrict__ out) {
  extern __shared__ float sm[];
  float* Tl   = sm;                 // 64 KB
  float* buf0 = sm + D * D;         // 32 KB
  float* buf1 = buf0 + 64 * D;      // 32 KB

  const int tid  = threadIdx.x;
  const int lane = tid & 31;
  const int wave = tid >> 5;
  const int l16  = lane & 15;
  const int kg   = (lane >> 4) * 2;
  const int n_slabs = (n_rows + 63) >> 6;

  // stage transform matrix once per block (first loop barrier publishes it)
  for (int p = tid; p < D * (D / 4); p += 256)
    *(v4f*)&Tl[p * 4] = *(const v4f*)&T[p * 4];

  const int m0    = (wave >> 1) * 16;
  const int nbase = (wave & 1) * 64;

  int s = blockIdx.x;
  if (s < n_slabs) stage64(emb, ids, n_rows, s << 6, buf0, tid);
  int par = 0;
  for (; s < n_slabs; s += gridDim.x) {
    async_wait();
    __syncthreads();
    float* cur = par ? buf1 : buf0;
    float* nxt = par ? buf0 : buf1;
    const int snext = s + (int)gridDim.x;
    if (snext < n_slabs) stage64(emb, ids, n_rows, snext << 6, nxt, tid);

    v8f acc[4] = {};
    for (int k0 = 0; k0 < D; k0 += 4) {
      v2f a;  // contiguous K pair per lane -> ds_load_b64
      a.x = cur[(m0 + l16) * D + k0 + kg];
      a.y = cur[(m0 + l16) * D + k0 + kg + 1];
#pragma unroll
      for (int n = 0; n < 4; ++n) {
        const int col = nbase + n * 16 + l16;
        v2f b;
        b.x = Tl[(k0 + kg) * D + col];
        b.y = Tl[(k0 + kg + 1) * D + col];
        acc[n] = __builtin_amdgcn_wmma_f32_16x16x4_f32(
            false, a, false, b, (short)0, acc[n], false, false);
      }
    }

    const int base = s << 6;
    const int row_half = (lane >> 4) * 8;
#pragma unroll
    for (int n = 0; n < 4; ++n) {
#pragma unroll
      for (int v = 0; v < 8; ++v) {
        const int grow = base + m0 + row_half + v;
        if (grow < n_rows)
          out[(size_t)grow * D + nbase + n * 16 + l16] = acc[n][v];
      }
    }
    par ^= 1;
  }
}

// ---------------------------------------------------------------------------
extern "C" void kernel_launch(void* const* d_in, const int* in_sizes, int n_in,
                              void* d_out, int out_size, void* d_ws, size_t ws_size,
                              hipStream_t stream) {
  const float* user_emb = (const float*)d_in[0];
  const float* item_emb = (const float*)d_in[1];
  const float* edge_emb = (const float*)d_in[2];
  const float* w_vec    = (const float*)d_in[3];
  const float* w_mat    = (const float*)d_in[4];
  const float* edge_w   = (const float*)d_in[5];
  const int*   user_ids = (const int*)d_in[6];
  const int*   item_ids = (const int*)d_in[7];
  const int U = in_sizes[6];
  const int I = in_sizes[7];

  float* Gx   = (float*)d_ws;
  float* Gy   = Gx + D * D;
  float* Aeff = Gy + D * D;
  float* Beff = Aeff + D * D;

  float* x_out  = (float*)d_out;
  float* y_out  = x_out + (size_t)U * D;
  float* es_out = y_out + (size_t)I * D;

  zero_kernel<<<(2 * D * D + 255) / 256, 256, 0, stream>>>(Gx, 2 * D * D);

  const size_t shmem_gram = (size_t)(2 * 64 * D) * sizeof(float);       // 64 KB
  gram_kernel<<<512, 256, shmem_gram, stream>>>(user_emb, user_ids, U, Gx);
  gram_kernel<<<512, 256, shmem_gram, stream>>>(item_emb, item_ids, I, Gy);

  build_kernel<<<1, D, 0, stream>>>(edge_emb, w_vec, w_mat, edge_w,
                                    Gx, Gy, Aeff, Beff, es_out);

  const size_t shmem_apply = (size_t)(D * D + 2 * 64 * D) * sizeof(float);  // 128 KB
  apply_kernel<<<1024, 256, shmem_apply, stream>>>(user_emb, user_ids, U, Aeff, x_out);
  apply_kernel<<<1024, 256, shmem_apply, stream>>>(item_emb, item_ids, I, Beff, y_out);
}